// Net_62998580298292
// MI455X (gfx1250) — compile-verified
//
#include <hip/hip_runtime.h>
#include <math.h>

typedef __attribute__((ext_vector_type(2))) float v2f;
typedef __attribute__((ext_vector_type(8))) float v8f;

#define FIN  512
#define FOUT 16

// ---------------- degree / normalization ----------------

__global__ void deg_init_kernel(float* __restrict__ deg, int n) {
  int i = blockIdx.x * blockDim.x + threadIdx.x;
  if (i < n) deg[i] = 1.0f;  // self loop contributes 1 to every node's degree
}

__global__ void deg_acc_kernel(const int* __restrict__ dst, float* __restrict__ deg, int E) {
  int e = blockIdx.x * blockDim.x + threadIdx.x;
  if (e < E) atomicAdd(&deg[dst[e]], 1.0f);
}

__global__ void deg_to_dis_kernel(float* __restrict__ deg, int n) {
  int i = blockIdx.x * blockDim.x + threadIdx.x;
  if (i < n) deg[i] = rsqrtf(deg[i]);  // deg >= 1 always (self loops)
}

// ---------------- layer-1 GEMM: h0 = x @ W1 (WMMA f32 16x16x4) ----------------
// One wave per 16-row tile. A (16x4 f32): lane l (0-15) holds row M=l; VGPR0 = K,
// VGPR1 = K+1 for lane-half 0, K+2/K+3 for lane-half 1. B (4x16): VGPR0 = row K
// (half 0) / row K+2 (half 1), VGPR1 = rows K+1 / K+3, column = lane&15.
// x (204.8 MB) is streamed once and exceeds the 192 MB L2 -> non-temporal loads
// so the edge index / feature buffers stay L2-resident.

__global__ void gemm1_wmma(const float* __restrict__ x, const float* __restrict__ W,
                           float* __restrict__ h, int n) {
  int wave = (int)(((size_t)blockIdx.x * blockDim.x + threadIdx.x) >> 5);
  int lane = threadIdx.x & 31;
  int row0 = wave * 16;
  if (row0 >= n) return;                 // whole wave exits together (EXEC all-1s for WMMA)
  int hh = lane >> 4;                    // lane half: 0 or 1
  int l  = lane & 15;
  int r_ld = row0 + l; if (r_ld >= n) r_ld = n - 1;
  const float* xrow = x + (size_t)r_ld * FIN;
  v8f acc = {};
#pragma unroll 8
  for (int k = 0; k < FIN; k += 4) {
    int ka = k + 2 * hh;
    v2f a = __builtin_nontemporal_load((const v2f*)(xrow + ka));   // TH=NT stream of x
    v2f b; b.x = W[ka * FOUT + l];      b.y = W[(ka + 1) * FOUT + l];
    acc = __builtin_amdgcn_wmma_f32_16x16x4_f32(false, a, false, b,
                                                (short)0, acc, false, false);
  }
  // D layout: VGPR r -> (M=r, N=l) for half 0, (M=r+8, N=l) for half 1
  float* hout = h + (size_t)(row0 + 8 * hh) * FOUT + l;
  if (row0 + 16 <= n) {                  // wave-uniform fast path: no per-store exec games
#pragma unroll
    for (int r = 0; r < 8; ++r) hout[r * FOUT] = acc[r];
  } else {
#pragma unroll
    for (int r = 0; r < 8; ++r) {
      int row = row0 + r + 8 * hh;
      if (row < n) h[(size_t)row * FOUT + l] = acc[r];
    }
  }
}

// ---------------- layer-2 GEMM: h2 = relu(h1) @ W2, K=16 ----------------

__global__ void gemm2_relu_wmma(const float* __restrict__ h1, const float* __restrict__ W,
                                float* __restrict__ h2, int n) {
  int wave = (int)(((size_t)blockIdx.x * blockDim.x + threadIdx.x) >> 5);
  int lane = threadIdx.x & 31;
  int row0 = wave * 16;
  if (row0 >= n) return;
  int hh = lane >> 4;
  int l  = lane & 15;
  int r_ld = row0 + l; if (r_ld >= n) r_ld = n - 1;
  const float* arow = h1 + (size_t)r_ld * FOUT;
  v8f acc = {};
#pragma unroll
  for (int k = 0; k < FOUT; k += 4) {
    int ka = k + 2 * hh;
    float a0 = arow[ka], a1 = arow[ka + 1];
    v2f a; a.x = fmaxf(a0, 0.0f);       a.y = fmaxf(a1, 0.0f);   // fused ReLU
    v2f b; b.x = W[ka * FOUT + l];      b.y = W[(ka + 1) * FOUT + l];
    acc = __builtin_amdgcn_wmma_f32_16x16x4_f32(false, a, false, b,
                                                (short)0, acc, false, false);
  }
  float* hout = h2 + (size_t)(row0 + 8 * hh) * FOUT + l;
  if (row0 + 16 <= n) {
#pragma unroll
    for (int r = 0; r < 8; ++r) hout[r * FOUT] = acc[r];
  } else {
#pragma unroll
    for (int r = 0; r < 8; ++r) {
      int row = row0 + r + 8 * hh;
      if (row < n) h2[(size_t)row * FOUT + l] = acc[r];
    }
  }
}

// ---------------- self-loop init: out[v] = h[v]*dis[v]^2 + bias ----------------

__global__ void init_out_kernel(const float* __restrict__ h, const float* __restrict__ dis,
                                const float* __restrict__ bias, float* __restrict__ out, int n) {
  size_t t = (size_t)blockIdx.x * blockDim.x + threadIdx.x;
  int c = (int)(t & 15);
  size_t row = t >> 4;
  if (row >= (size_t)n) return;
  float di = dis[row];
  out[row * FOUT + c] = h[row * FOUT + c] * di * di + bias[c];
}

// ---------------- edge scatter: out[d] += h[s] * dis[s]*dis[d] ----------------
// 16 lanes per edge (one per output column) -> coalesced 64B atomic bursts to L2.

__global__ void scatter_kernel(const int* __restrict__ src, const int* __restrict__ dst,
                               const float* __restrict__ dis, const float* __restrict__ h,
                               float* __restrict__ out, int E) {
  size_t t = (size_t)blockIdx.x * blockDim.x + threadIdx.x;
  int c = (int)(t & 15);
  size_t e = t >> 4;
  if (e >= (size_t)E) return;
  int s = src[e], d = dst[e];
  float nrm = dis[s] * dis[d];
  atomicAdd(&out[(size_t)d * FOUT + c], h[(size_t)s * FOUT + c] * nrm);
}

// ---------------- in-place log-softmax over 16 classes ----------------

__global__ void logsoftmax_kernel(float* __restrict__ out, int n) {
  int row = blockIdx.x * blockDim.x + threadIdx.x;
  if (row >= n) return;
  float* p = out + (size_t)row * FOUT;
  float v[FOUT];
  const float4* p4 = reinterpret_cast<const float4*>(p);
#pragma unroll
  for (int i = 0; i < 4; ++i) {
    float4 q = p4[i];
    v[4 * i + 0] = q.x; v[4 * i + 1] = q.y; v[4 * i + 2] = q.z; v[4 * i + 3] = q.w;
  }
  float m = v[0];
#pragma unroll
  for (int i = 1; i < FOUT; ++i) m = fmaxf(m, v[i]);
  float s = 0.0f;
#pragma unroll
  for (int i = 0; i < FOUT; ++i) s += expf(v[i] - m);
  float ls = m + logf(s);
  float4* w4 = reinterpret_cast<float4*>(p);
#pragma unroll
  for (int i = 0; i < 4; ++i) {
    float4 q;
    q.x = v[4 * i + 0] - ls; q.y = v[4 * i + 1] - ls;
    q.z = v[4 * i + 2] - ls; q.w = v[4 * i + 3] - ls;
    w4[i] = q;
  }
}

// ---------------- launch ----------------

extern "C" void kernel_launch(void* const* d_in, const int* in_sizes, int n_in,
                              void* d_out, int out_size, void* d_ws, size_t ws_size,
                              hipStream_t stream) {
  const float* x   = (const float*)d_in[0];
  const int*   ei  = (const int*)d_in[1];   // edge_index [2, E] int32
  const float* w1  = (const float*)d_in[2];
  const float* b1  = (const float*)d_in[3];
  const float* w2  = (const float*)d_in[4];
  const float* b2  = (const float*)d_in[5];

  int n = in_sizes[0] / FIN;       // 100000
  int E = in_sizes[1] / 2;         // 3200000
  const int* src = ei;
  const int* dst = ei + E;

  // workspace layout (256B aligned blocks)
  char* ws = (char*)d_ws;
  size_t off = 0;
  float* dis  = (float*)(ws + off); off += (((size_t)n * 4 + 255) / 256) * 256;
  float* h0   = (float*)(ws + off); off += (((size_t)n * FOUT * 4 + 255) / 256) * 256;
  float* out1 = (float*)(ws + off); off += (((size_t)n * FOUT * 4 + 255) / 256) * 256;
  float* h2   = (float*)(ws + off);
  float* out2 = (float*)d_out;     // accumulate logits directly in d_out

  const int B = 256;
  int gN   = (n + B - 1) / B;
  int gE   = (E + B - 1) / B;
  int gN16 = (int)(((size_t)n * 16 + B - 1) / B);
  int gE16 = (int)(((size_t)E * 16 + B - 1) / B);
  int gW   = (int)((((size_t)((n + 15) / 16)) * 32 + B - 1) / B);  // one wave per 16 rows

  // normalization: deg -> rsqrt(deg)   (dis buffer doubles as deg)
  deg_init_kernel<<<gN, B, 0, stream>>>(dis, n);
  deg_acc_kernel<<<gE, B, 0, stream>>>(dst, dis, E);
  deg_to_dis_kernel<<<gN, B, 0, stream>>>(dis, n);

  // layer 1
  gemm1_wmma<<<gW, B, 0, stream>>>(x, w1, h0, n);
  init_out_kernel<<<gN16, B, 0, stream>>>(h0, dis, b1, out1, n);
  scatter_kernel<<<gE16, B, 0, stream>>>(src, dst, dis, h0, out1, E);

  // layer 2 (ReLU fused into GEMM A-operand load)
  gemm2_relu_wmma<<<gW, B, 0, stream>>>(out1, w2, h2, n);
  init_out_kernel<<<gN16, B, 0, stream>>>(h2, dis, b2, out2, n);
  scatter_kernel<<<gE16, B, 0, stream>>>(src, dst, dis, h2, out2, E);

  // log-softmax in place on d_out
  logsoftmax_kernel<<<gN, B, 0, stream>>>(out2, n);
}